// SimpleGCNConv_22136261443775
// MI455X (gfx1250) — compile-verified
//
#include <hip/hip_runtime.h>

#define N_NODES 50000
#define N_EDGES 640000
#define D 128

typedef __attribute__((ext_vector_type(2))) float v2f;
typedef __attribute__((ext_vector_type(8))) float v8f;

#define FMAXV 3.402823466e38f

// nan_to_num, branchless: nan -> 0, +inf -> FLT_MAX, -inf -> -FLT_MAX.
// fmin/fmax clamp the infinities; NaN survives min/maxNum as +/-FMAX, so the
// final ordered-compare select (v_cmp_o + v_cndmask) zeroes it.
__device__ __forceinline__ float fix_val(float v) {
    float r = fminf(fmaxf(v, -FMAXV), FMAXV);
    return (v == v) ? r : 0.0f;
}

// edge weights, branchless: nan/±inf -> 0, else |v|.
// |v| <= FMAX is false for both NaN and inf -> single cmp + cndmask.
__device__ __forceinline__ float fix_ew(float v) {
    float a = fabsf(v);
    return (a <= FMAXV) ? a : 0.0f;
}

__global__ void gcn_zero(float* __restrict__ p, int n) {
    int i = blockIdx.x * blockDim.x + threadIdx.x;
    if (i < n) p[i] = 0.0f;
}

// One wave32 per edge; each lane handles 4 consecutive features (float4).
// Gather x[src], scrub, scale by |w|, atomic-add into acc[dst]; lane0 bumps deg.
// Whole working set (x 25.6MB, acc 25.6MB, edges 10MB) is L2-resident on the
// 192MB L2, so this phase is bounded by L2 atomic throughput, not HBM.
__global__ void gcn_scatter(const float* __restrict__ x,
                            const long long* __restrict__ ei,
                            const float* __restrict__ ew,
                            float* __restrict__ acc,
                            float* __restrict__ deg) {
    unsigned gid  = blockIdx.x * blockDim.x + threadIdx.x;
    unsigned e    = gid >> 5;
    unsigned lane = gid & 31u;
    if (e >= N_EDGES) return;

    long long dst = ei[e];            // edge_index[0, e]
    long long src = ei[N_EDGES + e];  // edge_index[1, e]
    float w = fix_ew(ew[e]);

    const float4* xr = (const float4*)(x + (size_t)src * D);
    float4 v = xr[lane];
    v.x = fix_val(v.x) * w;
    v.y = fix_val(v.y) * w;
    v.z = fix_val(v.z) * w;
    v.w = fix_val(v.w) * w;

    float* ar = acc + (size_t)dst * D + lane * 4;
    atomicAdd(ar + 0, v.x);
    atomicAdd(ar + 1, v.y);
    atomicAdd(ar + 2, v.z);
    atomicAdd(ar + 3, v.w);
    if (lane == 0) atomicAdd(deg + dst, 1.0f);
}

// Fused normalize + GEMM with V_WMMA_F32_16X16X4_F32 (exact fp32).
// Block = 8 waves; wave w computes the 16x16 tile (rows = 16 nodes of this
// block, cols = output features [16w, 16w+16)). K loop: 32 steps of 4.
// A layout (16x4 f32): lane&15 = row M, lane>>4 selects K pair {0,1}/{2,3}.
// B layout (4x16 f32): lane&15 = col N, same K-pair split.
// C/D layout: 8 VGPRs, N = lane&15, M = i + 8*(lane>=16).
// Grid is exact (50000 % 16 == 0), no guards -> EXEC stays all-1s for WMMA.
__global__ void __launch_bounds__(256)
gcn_gemm(const float* __restrict__ acc,
         const float* __restrict__ deg,
         const float* __restrict__ W,
         const float* __restrict__ bias,
         float* __restrict__ out) {
    const int lane   = threadIdx.x & 31;
    const int wave   = threadIdx.x >> 5;
    const int m_base = blockIdx.x * 16;
    const int o_base = wave * 16;
    const int nsub   = lane & 15;  // row M for A, col N for B/C/D
    const int kgrp   = lane >> 4;  // which K pair

    const int   node = m_base + nsub;
    const float dinv = 1.0f / fmaxf(deg[node], 1.0f);

    const v2f* aRow = (const v2f*)(acc + (size_t)node * D);
    const v2f* wRow = (const v2f*)(W + (size_t)(o_base + nsub) * D);

    const float bval = bias[o_base + nsub];
    v8f c;
#pragma unroll
    for (int i = 0; i < 8; ++i) c[i] = bval;

#pragma unroll
    for (int kb = 0; kb < D; kb += 4) {
        v2f av = aRow[(kb >> 1) + kgrp];
        v2f a;
        a.x = fix_val(av.x * dinv);
        a.y = fix_val(av.y * dinv);
        v2f b = wRow[(kb >> 1) + kgrp];
        // D = A(16x4) * B(4x16) + C  — fp32 WMMA, exact precision
        c = __builtin_amdgcn_wmma_f32_16x16x4_f32(
                /*neg_a=*/false, a, /*neg_b=*/false, b,
                /*c_mod=*/(short)0, c, /*reuse_a=*/false, /*reuse_b=*/false);
    }

    float* o = out + (size_t)m_base * D + o_base + nsub;
#pragma unroll
    for (int i = 0; i < 8; ++i) {
        int mm = (lane < 16) ? i : (i + 8);
        o[(size_t)mm * D] = c[i];
    }
}

extern "C" void kernel_launch(void* const* d_in, const int* in_sizes, int n_in,
                              void* d_out, int out_size, void* d_ws, size_t ws_size,
                              hipStream_t stream) {
    const float*     x  = (const float*)d_in[0];
    const long long* ei = (const long long*)d_in[1];  // int64 edge_index [2, E]
    const float*     ew = (const float*)d_in[2];
    const float*     W  = (const float*)d_in[3];
    const float*     b  = (const float*)d_in[4];
    float* out = (float*)d_out;

    float* acc = (float*)d_ws;                 // [N_NODES, D]
    float* deg = acc + (size_t)N_NODES * D;    // [N_NODES]

    const int nz = N_NODES * D + N_NODES;
    gcn_zero<<<(nz + 255) / 256, 256, 0, stream>>>(acc, nz);

    const long long scatter_threads = (long long)N_EDGES * 32;
    gcn_scatter<<<(unsigned)((scatter_threads + 255) / 256), 256, 0, stream>>>(
        x, ei, ew, acc, deg);

    gcn_gemm<<<N_NODES / 16, 256, 0, stream>>>(acc, deg, W, b, out);
}